// EnhancedGMBLayer_47605417509004
// MI455X (gfx1250) — compile-verified
//
#include <hip/hip_runtime.h>
#include <hip/hip_bf16.h>

// ---------------- problem constants ----------------
#define NN      8192
#define GG      16
#define MAXL    768
#define DD      128
#define DIN     256     // EXPAND * D
#define DSTATE  16
#define DCONV   4
#define DTRANK  8
#define NSLOT   (GG * MAXL)   // 12288
#define DBCP    48            // dbc row stride (40 valid, padded to 48)

typedef __attribute__((ext_vector_type(2))) float v2f;
typedef __attribute__((ext_vector_type(8))) float v8f;

// ---------------- zero workspace region ----------------
__global__ __launch_bounds__(256) void gmb_zero_kernel(float* __restrict__ p, int n) {
    for (int i = blockIdx.x * blockDim.x + threadIdx.x; i < n; i += gridDim.x * blockDim.x)
        p[i] = 0.0f;
}

// ---------------- stable rank by batch (one wave per graph) ----------------
__global__ __launch_bounds__(32) void gmb_rank_kernel(const int* __restrict__ batch,
                                                      int* __restrict__ n2s) {
    const int g = blockIdx.x;
    const int lane = threadIdx.x;
    int c = 0;
    for (int base = 0; base < NN; base += 32) {
        const int i = base + lane;
        const bool m = (batch[i] == g);
        const unsigned int bal = (unsigned int)__ballot(m);
        if (m) n2s[i] = g * MAXL + c + __popc(bal & ((1u << lane) - 1u));
        c += __popc(bal);
    }
}

// ---------------- pack dense (fwd) + flipped dense (bwd) ----------------
__global__ __launch_bounds__(128) void gmb_pack_kernel(const float* __restrict__ x,
                                                       const int* __restrict__ n2s,
                                                       float* __restrict__ df,
                                                       float* __restrict__ db) {
    const int node = blockIdx.x;
    const int t = threadIdx.x;
    const int slot = n2s[node];
    const float v = x[(size_t)node * DD + t];
    df[(size_t)slot * DD + t] = v;
    const int g = slot / MAXL;
    const int l = slot % MAXL;
    db[(size_t)(g * MAXL + (MAXL - 1 - l)) * DD + t] = v;
}

// ---------------- pad xp_w (40x256) -> wpad (48x256, zero rows 40..47) ----------------
__global__ __launch_bounds__(256) void gmb_padw_kernel(const float* __restrict__ w,
                                                       float* __restrict__ wpad) {
    const int row = blockIdx.x;       // 0..47
    const int col = threadIdx.x;      // 0..255
    wpad[(size_t)row * DIN + col] = (row < (DTRANK + 2 * DSTATE))
                                        ? w[(size_t)row * DIN + col] : 0.0f;
}

// ---------------- fp32 WMMA GEMM: C[M, NT*16] tile = A[M,K] * W[N,K]^T ----------------
// One wave computes a 16 x (NT*16) tile with NT independent accumulators.
// Software-pipelined: next k-step's A/B fragments are loaded (double-buffered
// in registers) before the WMMAs consuming the current fragments, so
// s_wait_loadcnt sinks past the matrix ops. EXEC stays all-ones throughout.
template <int NT>
__global__ __launch_bounds__(32) void gmb_wmma_gemm_nt(const float* __restrict__ A,
                                                       const float* __restrict__ W,
                                                       float* __restrict__ C,
                                                       int K, int ldc) {
    const int lane = threadIdx.x;
    const int m0 = blockIdx.x * 16;
    const int n0 = blockIdx.y * (16 * NT);
    const int row = lane & 15;              // M (for A) / N (for B)
    const int kg  = (lane >> 4) * 2;        // lanes 0-15: K=0,1 ; lanes 16-31: K=2,3

    const float* aBase = A + (size_t)(m0 + row) * K + kg;
    const float* wBase[NT];
#pragma unroll
    for (int t = 0; t < NT; ++t) {
        wBase[t] = W + (size_t)(n0 + t * 16 + row) * K + kg;
        __builtin_prefetch(wBase[t], 0, 3);   // in-bounds weight-row prefetch
    }

    v8f acc[NT];
#pragma unroll
    for (int t = 0; t < NT; ++t) acc[t] = (v8f){0.f, 0.f, 0.f, 0.f, 0.f, 0.f, 0.f, 0.f};

    // prologue: load first fragments
    v2f a_cur = *reinterpret_cast<const v2f*>(aBase);
    v2f b_cur[NT];
#pragma unroll
    for (int t = 0; t < NT; ++t) b_cur[t] = *reinterpret_cast<const v2f*>(wBase[t]);

    // steady state: prefetch next fragments, then consume current ones
#pragma unroll 2
    for (int kb = 0; kb < K - 4; kb += 4) {
        const v2f a_nxt = *reinterpret_cast<const v2f*>(aBase + kb + 4);
        v2f b_nxt[NT];
#pragma unroll
        for (int t = 0; t < NT; ++t)
            b_nxt[t] = *reinterpret_cast<const v2f*>(wBase[t] + kb + 4);

#pragma unroll
        for (int t = 0; t < NT; ++t)
            acc[t] = __builtin_amdgcn_wmma_f32_16x16x4_f32(
                false, a_cur, false, b_cur[t], (short)0, acc[t], false, false);

        a_cur = a_nxt;
#pragma unroll
        for (int t = 0; t < NT; ++t) b_cur[t] = b_nxt[t];
    }

    // epilogue: last k-step
#pragma unroll
    for (int t = 0; t < NT; ++t)
        acc[t] = __builtin_amdgcn_wmma_f32_16x16x4_f32(
            false, a_cur, false, b_cur[t], (short)0, acc[t], false, false);

    const int mh = (lane >> 4) * 8;         // D layout: lanes 0-15 -> M=0..7, 16-31 -> M=8..15
#pragma unroll
    for (int t = 0; t < NT; ++t) {
        const int col = n0 + t * 16 + row;
#pragma unroll
        for (int r = 0; r < 8; ++r)
            C[(size_t)(m0 + mh + r) * ldc + col] = acc[t][r];
    }
}

// ---------------- causal depthwise conv (K=4) + bias + SiLU ----------------
__global__ __launch_bounds__(256) void gmb_conv_silu_kernel(const float* __restrict__ xz,
                                                            const float* __restrict__ cw,
                                                            const float* __restrict__ cb,
                                                            float* __restrict__ xc) {
    const int slot = blockIdx.x;
    const int d = threadIdx.x;
    const int l = slot % MAXL;
    float acc = cb[d];
#pragma unroll
    for (int j = 0; j < DCONV; ++j) {
        const int ll = l - (DCONV - 1) + j;
        if (ll >= 0)
            acc += xz[(size_t)(slot - (DCONV - 1) + j) * (2 * DIN) + d] * cw[d * DCONV + j];
    }
    xc[(size_t)slot * DIN + d] = acc / (1.0f + __expf(-acc));   // silu
}

// ---------------- fused dt/softplus + selective scan + D-skip + z-gate ----------------
// grid (G, DIN/64), block 64: one thread per channel d, 16 states in registers.
__global__ __launch_bounds__(64) void gmb_scan_kernel(const float* __restrict__ dbc,
                                                      float* __restrict__ xcy,   // in: xc, out: y
                                                      const float* __restrict__ xz, // z = cols [256,512)
                                                      const float* __restrict__ dt_w,
                                                      const float* __restrict__ dt_b,
                                                      const float* __restrict__ A_log,
                                                      const float* __restrict__ Dp) {
    const int g = blockIdx.x;
    const int d = blockIdx.y * 64 + threadIdx.x;
    const int tid = threadIdx.x;

    float dtw[DTRANK];
#pragma unroll
    for (int r = 0; r < DTRANK; ++r) dtw[r] = dt_w[d * DTRANK + r];
    float an[DSTATE];
#pragma unroll
    for (int s = 0; s < DSTATE; ++s) an[s] = -__expf(A_log[d * DSTATE + s]); // A = -exp(A_log)
    const float dtb = dt_b[d];
    const float dpar = Dp[d];

    float h[DSTATE];
#pragma unroll
    for (int s = 0; s < DSTATE; ++s) h[s] = 0.0f;

    __shared__ float lbc[DTRANK + 2 * DSTATE];   // 40: [0,8)=rank, [8,24)=B, [24,40)=C

    for (int l = 0; l < MAXL; ++l) {
        const int slot = g * MAXL + l;
        if (tid < (DTRANK + 2 * DSTATE)) lbc[tid] = dbc[(size_t)slot * DBCP + tid];
        __syncthreads();

        float dtv = dtb;
#pragma unroll
        for (int r = 0; r < DTRANK; ++r) dtv += lbc[r] * dtw[r];
        dtv = (dtv > 20.0f) ? dtv : log1pf(__expf(dtv));   // softplus

        const float xcv = xcy[(size_t)slot * DIN + d];
        const float zv  = xz[(size_t)slot * (2 * DIN) + DIN + d];
        const float dtx = dtv * xcv;

        float acc = 0.0f;
#pragma unroll
        for (int s = 0; s < DSTATE; ++s) {
            const float da = __expf(dtv * an[s]);
            h[s] = da * h[s] + dtx * lbc[DTRANK + s];
            acc += h[s] * lbc[DTRANK + DSTATE + s];
        }
        float y = acc + dpar * xcv;
        y *= zv / (1.0f + __expf(-zv));                    // * silu(z)
        __syncthreads();                                    // lbc reuse next iter
        xcy[(size_t)slot * DIN + d] = y;
    }
}

// ---------------- gate + combine + scatter to original node order ----------------
__global__ __launch_bounds__(256) void gmb_gate_kernel(const float* __restrict__ of,
                                                       const float* __restrict__ ob,
                                                       const int* __restrict__ n2s,
                                                       const float* __restrict__ gw,
                                                       const float* __restrict__ gb,
                                                       float* __restrict__ out) {
    __shared__ float fv[8 * 256];   // per node: [0,128)=fwd, [128,256)=bwd
    const int tid = threadIdx.x;
    const int ni = tid >> 5;
    const int lane = tid & 31;
    const int nodeBase = blockIdx.x * 8;

    for (int nn = 0; nn < 8; ++nn) {
        const int node = nodeBase + nn;
        const int slot = n2s[node];
        const int g = slot / MAXL, l = slot % MAXL;
        const int bslot = g * MAXL + (MAXL - 1 - l);
        const float v = (tid < DD) ? of[(size_t)slot * DD + tid]
                                   : ob[(size_t)bslot * DD + (tid - DD)];
        fv[nn * 256 + tid] = v;
    }
    __syncthreads();

    const int node = nodeBase + ni;
    const float* f = fv + ni * 256;

    float a0 = gb[lane], a1 = gb[lane + 32], a2 = gb[lane + 64], a3 = gb[lane + 96];
    const float* gw0 = gw + (size_t)(lane)       * 256;
    const float* gw1 = gw + (size_t)(lane + 32)  * 256;
    const float* gw2 = gw + (size_t)(lane + 64)  * 256;
    const float* gw3 = gw + (size_t)(lane + 96)  * 256;
#pragma unroll 4
    for (int k = 0; k < 256; ++k) {
        const float fk = f[k];
        a0 += fk * gw0[k];
        a1 += fk * gw1[k];
        a2 += fk * gw2[k];
        a3 += fk * gw3[k];
    }
    float accs[4] = {a0, a1, a2, a3};
#pragma unroll
    for (int o4 = 0; o4 < 4; ++o4) {
        const int o = lane + o4 * 32;
        const float s = 1.0f / (1.0f + __expf(-accs[o4]));
        out[(size_t)node * DD + o] = s * f[o] + (1.0f - s) * f[DD + o];
    }
}

// ---------------- host orchestration ----------------
struct MambaParams {
    const float *in_w, *conv_w, *conv_b, *xp_w, *dt_w, *dt_b, *A_log, *Dp, *out_w;
};

extern "C" void kernel_launch(void* const* d_in, const int* in_sizes, int n_in,
                              void* d_out, int out_size, void* d_ws, size_t ws_size,
                              hipStream_t stream) {
    const float* x     = (const float*)d_in[0];
    const int*   batch = (const int*)d_in[3];

    auto getp = [&](int base) {
        MambaParams p;
        p.in_w   = (const float*)d_in[base + 0];
        p.conv_w = (const float*)d_in[base + 1];
        p.conv_b = (const float*)d_in[base + 2];
        p.xp_w   = (const float*)d_in[base + 3];
        p.dt_w   = (const float*)d_in[base + 4];
        p.dt_b   = (const float*)d_in[base + 5];
        p.A_log  = (const float*)d_in[base + 6];
        p.Dp     = (const float*)d_in[base + 7];
        p.out_w  = (const float*)d_in[base + 8];
        return p;
    };
    MambaParams fm = getp(4);
    MambaParams bm = getp(13);
    const float* gate_w = (const float*)d_in[22];
    const float* gate_b = (const float*)d_in[23];

    // ---- workspace carve-out ----
    int*   n2s  = (int*)d_ws;
    float* base = (float*)((char*)d_ws + 65536);
    const size_t SZ_DENSE = (size_t)NSLOT * DD;       // 1,572,864
    const size_t SZ_XZ    = (size_t)NSLOT * 2 * DIN;  // 6,291,456
    const size_t SZ_XC    = (size_t)NSLOT * DIN;      // 3,145,728
    const size_t SZ_DBC   = (size_t)NSLOT * DBCP;     //   589,824
    float* dense_f = base;
    float* dense_b = dense_f + SZ_DENSE;
    float* xz      = dense_b + SZ_DENSE;
    float* xcy     = xz + SZ_XZ;
    float* dbc     = xcy + SZ_XC;
    float* out_f   = dbc + SZ_DBC;
    float* out_b   = out_f + SZ_DENSE;
    float* wpad    = out_b + SZ_DENSE;                // 48*256 floats

    // zero padded dense buffers (fwd + bwd are contiguous)
    gmb_zero_kernel<<<2048, 256, 0, stream>>>(dense_f, (int)(2 * SZ_DENSE));
    // stable sort by batch -> slot per node
    gmb_rank_kernel<<<GG, 32, 0, stream>>>(batch, n2s);
    // pack dense + flipped dense
    gmb_pack_kernel<<<NN, 128, 0, stream>>>(x, n2s, dense_f, dense_b);

    for (int dir = 0; dir < 2; ++dir) {
        const MambaParams& p = dir ? bm : fm;
        const float* din  = dir ? dense_b : dense_f;
        float*       outd = dir ? out_b : out_f;

        // xz = dense @ in_w^T   (K=128, N=512): 16x64 tiles per wave
        gmb_wmma_gemm_nt<4><<<dim3(NSLOT / 16, 512 / 64), 32, 0, stream>>>(
            din, p.in_w, xz, DD, 2 * DIN);
        // depthwise causal conv + silu -> xc
        gmb_conv_silu_kernel<<<NSLOT, DIN, 0, stream>>>(xz, p.conv_w, p.conv_b, xcy);
        // pad xp_w to 48 rows (zeros beyond 40) so the GEMM is branch-free
        gmb_padw_kernel<<<DBCP, DIN, 0, stream>>>(p.xp_w, wpad);
        // dbc = xc @ wpad^T     (K=256, N=48 padded): 16x48 tiles per wave
        gmb_wmma_gemm_nt<3><<<dim3(NSLOT / 16, 1), 32, 0, stream>>>(
            xcy, wpad, dbc, DIN, DBCP);
        // fused dt + selective scan + skip + z-gate (y overwrites xc)
        gmb_scan_kernel<<<dim3(GG, DIN / 64), 64, 0, stream>>>(
            dbc, xcy, xz, p.dt_w, p.dt_b, p.A_log, p.Dp);
        // out = y @ out_w^T     (K=256, N=128): 16x64 tiles per wave
        gmb_wmma_gemm_nt<4><<<dim3(NSLOT / 16, 128 / 64), 32, 0, stream>>>(
            xcy, p.out_w, outd, DIN, DD);
    }

    // gate combine + scatter back to original node order
    gmb_gate_kernel<<<NN / 8, 256, 0, stream>>>(
        out_f, out_b, n2s, gate_w, gate_b, (float*)d_out);
}